// PositionEmbedding_84318797955295
// MI455X (gfx1250) — compile-verified
//
#include <hip/hip_runtime.h>
#include <math.h>

// Problem constants (from reference)
#define NUM_HEADS   32
#define NUM_BUCKETS 32
#define TSTRIDE     4096      // padded length of per-head diagonal table
#define TK          16        // k-rows per block in the streaming kernel
#define BLOCK       256       // 8 waves of 32

typedef float v4f __attribute__((ext_vector_type(4)));
typedef int   v4i __attribute__((vector_size(16)));

#if defined(__AMDGCN__) && __has_builtin(__builtin_amdgcn_global_load_async_to_lds_b128)
#define USE_ASYNC_LDS 1
typedef __attribute__((address_space(1))) v4i* gaddr_v4i;
typedef __attribute__((address_space(3))) v4i* laddr_v4i;
#else
#define USE_ASYNC_LDS 0
#endif

// ---------------------------------------------------------------------------
// Kernel B (placed first so the disasm snippet shows it): stream 512 MiB.
// Block = (head h, k-tile of TK rows). LDS holds T[h][0..4095] (16 KB),
// filled via CDNA5 async global->LDS loads (ASYNCcnt, s_wait_asynccnt).
// Each thread then writes float4 columns with nontemporal b128 stores:
//   out[h,k,q] = lds[ q + (K-1) - k ]   (contiguous in q per row).
// ---------------------------------------------------------------------------
__global__ void __launch_bounds__(BLOCK)
stream_bias(const float* __restrict__ T, float* __restrict__ out,
            int K, int Q) {
    __shared__ float l[TSTRIDE];        // 16 KB

    const int h  = blockIdx.y;
    const int k0 = blockIdx.x * TK;
    const float* Trow = T + (size_t)h * TSTRIDE;

    // ---- Stage table row into LDS -----------------------------------------
#if USE_ASYNC_LDS
    // 4096 floats = 256 lanes * 4 iters * one async b128 each
    #pragma unroll
    for (int it = 0; it < TSTRIDE / (BLOCK * 4); ++it) {
        int idx = (it * BLOCK + threadIdx.x) * 4;
        __builtin_amdgcn_global_load_async_to_lds_b128(
            (gaddr_v4i)(Trow + idx),
            (laddr_v4i)(&l[idx]),
            0 /*imm offset*/, 0 /*cpol*/);
    }
  #if __has_builtin(__builtin_amdgcn_s_wait_asynccnt)
    __builtin_amdgcn_s_wait_asynccnt(0);
  #else
    asm volatile("s_wait_asynccnt 0" ::: "memory");
  #endif
#else
    #pragma unroll
    for (int it = 0; it < TSTRIDE / (BLOCK * 4); ++it) {
        int idx = (it * BLOCK + threadIdx.x) * 4;
        v4f v = *(const v4f*)(Trow + idx);
        *(v4f*)(&l[idx]) = v;
    }
#endif
    __syncthreads();

    // ---- Stream output ----------------------------------------------------
    const int QV4 = Q >> 2;             // float4 columns per row (512)
    v4f* outv = (v4f*)(out + ((size_t)h * K + k0) * Q);
    const float* __restrict__ lr = l;

    #pragma unroll
    for (int r = 0; r < TK; ++r) {
        int k = k0 + r;
        if (k >= K) break;
        int base = (K - 1) - k;         // lds index of q = 0
        for (int v = threadIdx.x; v < QV4; v += BLOCK) {
            int j = (v << 2) + base;    // q + (K-1) - k
            v4f val;
            val.x = lr[j + 0];
            val.y = lr[j + 1];
            val.z = lr[j + 2];
            val.w = lr[j + 3];
#if defined(__AMDGCN__)
            __builtin_nontemporal_store(val, &outv[(size_t)r * QV4 + v]);
#else
            outv[(size_t)r * QV4 + v] = val;
#endif
        }
    }
}

// ---------------------------------------------------------------------------
// Kernel A: build flipped diagonal table.
//   T[h][j] = weight[h, bucket(d)]  with d = (K-1) - j,  j in [0, TSTRIDE)
// T5 bidirectional bucketing: 32 buckets, max_distance=128.
// ---------------------------------------------------------------------------
__global__ void build_diag_table(const float* __restrict__ weight,
                                 float* __restrict__ T, int K) {
    const int h = blockIdx.x;
    const float inv_log16 = 0.36067376022224085f; // 1/ln(16)
    for (int j = threadIdx.x; j < TSTRIDE; j += blockDim.x) {
        int d = (K - 1) - j;            // rel = k - q
        int ret = 0;
        int a;
        if (d > 0) { ret = NUM_BUCKETS / 2; a = d; } else { a = -d; }
        int b;
        if (a < 8) {                    // max_exact = 8
            b = a;
        } else {
            // 8 + int( log(a/8)/log(16) * 8 ), clamped to 15
            float lf = logf((float)a * 0.125f) * inv_log16 * 8.0f;
            b = 8 + (int)lf;            // trunc toward zero, a>=8 -> lf>=0
            if (b > 15) b = 15;
        }
        int bucket = ret + b;
        T[(size_t)h * TSTRIDE + j] = weight[h * NUM_BUCKETS + bucket];
    }
}

// ---------------------------------------------------------------------------
// Host launcher
// ---------------------------------------------------------------------------
extern "C" void kernel_launch(void* const* d_in, const int* in_sizes, int n_in,
                              void* d_out, int out_size, void* d_ws, size_t ws_size,
                              hipStream_t stream) {
    const float* weight = (const float*)d_in[0];
    float* out = (float*)d_out;

    // out_size = NUM_HEADS * K * Q with K == Q (reference: 2048 x 2048)
    long long kq = (long long)out_size / NUM_HEADS;
    int side = (int)llround(sqrt((double)kq));
    int K = side, Q = side;

    float* T = (float*)d_ws;            // NUM_HEADS * TSTRIDE floats = 512 KB

    build_diag_table<<<NUM_HEADS, 256, 0, stream>>>(weight, T, K);

    dim3 grid((K + TK - 1) / TK, NUM_HEADS);
    stream_bias<<<grid, BLOCK, 0, stream>>>(T, out, K, Q);
}